// SpatialRefinementGAT_57037165691746
// MI455X (gfx1250) — compile-verified
//
#include <hip/hip_runtime.h>
#include <hip/hip_bf16.h>
#include <math.h>
#include <float.h>

typedef __attribute__((ext_vector_type(2))) float v2f;
typedef __attribute__((ext_vector_type(8))) float v8f;
typedef __attribute__((ext_vector_type(4))) unsigned int u32x4;
typedef __attribute__((ext_vector_type(8))) int i32x8;
typedef __attribute__((ext_vector_type(4))) int i32x4;

#define GAT_N 100000
#define GAT_E 3200000
#define GAT_ETOT (GAT_E + GAT_N)
#define NEG_SLOPE 0.2f

static __device__ __forceinline__ float gat_leaky(float v) {
  return v > 0.0f ? v : NEG_SLOPE * v;
}

static __device__ __forceinline__ void atomicMaxF(float* addr, float val) {
  int* ia = (int*)addr;
  int cur = __float_as_int(*addr);
  while (__int_as_float(cur) < val) {
    int old = atomicCAS(ia, cur, __float_as_int(val));
    if (old == cur) break;
    cur = old;
  }
}

static __device__ __forceinline__ void edge_nodes(const int* __restrict__ ei,
                                                  int e, int& s, int& d) {
  if (e < GAT_E) { s = ei[e]; d = ei[GAT_E + e]; }
  else           { s = e - GAT_E; d = s; }          // self-loop
}

// ---------------------------------------------------------------------------
// GEMM: [Nrows x K] @ [K x 64] -> [Nrows x 64] using V_WMMA_F32_16X16X4_F32.
// K is a compile-time multiple of 4 (16 or 64). W tile (K x 64 f32) is staged
// into LDS by the Tensor Data Mover (tensor_load_to_lds, TENSORcnt), then one
// wave computes a 16x64 slab with 4 f32 accumulators. Row index is clamped so
// all A loads are unconditional global_load_b64; stores take a wave-uniform
// unguarded fast path except in the single boundary wave.
// ---------------------------------------------------------------------------
template <int K>
__global__ void gat_gemm_wmma64(const float* __restrict__ A,
                                const float* __restrict__ W,
                                float* __restrict__ Hout,
                                int Nrows) {
  __shared__ float sW[K * 64];            // only LDS object -> LDS offset 0

  if (threadIdx.x < 32) {                 // one wave issues the TDM transfer
    unsigned long long ga = (unsigned long long)W;
    u32x4 g0;
    g0[0] = 1u;                                              // count=1, user D#
    g0[1] = 0u;                                              // lds_addr = 0
    g0[2] = (unsigned int)(ga & 0xFFFFFFFFu);                // global_addr lo
    g0[3] = ((unsigned int)(ga >> 32) & 0x01FFFFFFu)         // global_addr hi
            | 0x80000000u;                                   // type=2 (image)
    i32x8 g1;
    g1[0] = 0x00020000;                  // workgroup_mask=0, data_size=4B
    g1[1] = (64 & 0xFFFF) << 16;         // tensor_dim0 = 64 (bits 79:48)
    g1[2] = (K & 0xFFFF) << 16;          // tensor_dim0 hi=0 | tensor_dim1 = K
    g1[3] = 64 << 16;                    // tensor_dim1 hi=0 | tile_dim0 = 64
    g1[4] = K & 0xFFFF;                  // tile_dim1 = K, tile_dim2 = 0
    g1[5] = 64;                          // tensor_dim0_stride = 64 elements
    g1[6] = 0;
    g1[7] = 0;                           // tensor_dim1_stride unused (2D tile)
    i32x4 z4 = {0, 0, 0, 0};
    i32x8 z8 = {0, 0, 0, 0, 0, 0, 0, 0};
    __builtin_amdgcn_tensor_load_to_lds(g0, g1, z4, z4, z8, 0);
    __builtin_amdgcn_s_wait_tensorcnt(0);
  }
  __syncthreads();

  const int wave = threadIdx.x >> 5;
  const int lane = threadIdx.x & 31;
  const int row0 = (blockIdx.x * 8 + wave) << 4;   // 16 rows per wave
  const int m    = lane & 15;
  const int ko   = (lane >> 4) << 1;               // 0 (lanes 0-15) or 2
  const int n    = lane & 15;
  int r = row0 + m;
  if (r > Nrows - 1) r = Nrows - 1;                // clamp: loads always safe
  const float* arow = A + (long)r * K + ko;

  v8f acc0 = {0,0,0,0,0,0,0,0};
  v8f acc1 = {0,0,0,0,0,0,0,0};
  v8f acc2 = {0,0,0,0,0,0,0,0};
  v8f acc3 = {0,0,0,0,0,0,0,0};

#pragma unroll
  for (int k = 0; k < K; k += 4) {
    v2f a = *(const v2f*)(arow + k);               // unconditional b64 load

    const float* wr0 = &sW[(k + ko) * 64];
    const float* wr1 = wr0 + 64;
    v2f b0, b1, b2, b3;
    b0.x = wr0[n];      b0.y = wr1[n];
    b1.x = wr0[n + 16]; b1.y = wr1[n + 16];
    b2.x = wr0[n + 32]; b2.y = wr1[n + 32];
    b3.x = wr0[n + 48]; b3.y = wr1[n + 48];

    acc0 = __builtin_amdgcn_wmma_f32_16x16x4_f32(false, a, false, b0, (short)0, acc0, false, false);
    acc1 = __builtin_amdgcn_wmma_f32_16x16x4_f32(false, a, false, b1, (short)0, acc1, false, false);
    acc2 = __builtin_amdgcn_wmma_f32_16x16x4_f32(false, a, false, b2, (short)0, acc2, false, false);
    acc3 = __builtin_amdgcn_wmma_f32_16x16x4_f32(false, a, false, b3, (short)0, acc3, false, false);
  }

  const int rowadd = (lane >> 4) << 3;             // +8 rows for upper half
  if (row0 + 16 <= Nrows) {
    // Fast path (all but the final partial wave): no per-row guards.
#pragma unroll
    for (int i = 0; i < 8; i++) {
      float* o = &Hout[(long)(row0 + i + rowadd) * 64 + n];
      o[0]  = acc0[i];
      o[16] = acc1[i];
      o[32] = acc2[i];
      o[48] = acc3[i];
    }
  } else {
#pragma unroll
    for (int i = 0; i < 8; i++) {
      int rr = row0 + i + rowadd;
      if (rr < Nrows) {
        float* o = &Hout[(long)rr * 64 + n];
        o[0]  = acc0[i];
        o[16] = acc1[i];
        o[32] = acc2[i];
        o[48] = acc3[i];
      }
    }
  }
}

// Zero-pad x [N,10] -> XP [N,16]
__global__ void gat_pad_x(const float* __restrict__ x, float* __restrict__ XP, int Nn) {
  int i = blockIdx.x * blockDim.x + threadIdx.x;
  if (i >= Nn * 16) return;
  int nrow = i >> 4, j = i & 15;
  XP[i] = (j < 10) ? x[nrow * 10 + j] : 0.0f;
}

// Zero-pad W1 [10,64] -> WP [16,64]
__global__ void gat_pad_w1(const float* __restrict__ W1, float* __restrict__ WP) {
  int i = blockIdx.x * blockDim.x + threadIdx.x;
  if (i >= 16 * 64) return;
  WP[i] = ((i >> 6) < 10) ? W1[i] : 0.0f;
}

// Plain GEMM for layer 3: [N x 64] @ [64 x 4]
__global__ void gat_gemm3(const float* __restrict__ A, const float* __restrict__ W,
                          float* __restrict__ Hout, int Nrows) {
  int nrow = blockIdx.x * blockDim.x + threadIdx.x;
  if (nrow >= Nrows) return;
  float acc[4] = {0.f, 0.f, 0.f, 0.f};
  const float* a = &A[(long)nrow * 64];
#pragma unroll 8
  for (int k = 0; k < 64; k++) {
    float av = a[k];
    acc[0] += av * W[k * 4 + 0];
    acc[1] += av * W[k * 4 + 1];
    acc[2] += av * W[k * 4 + 2];
    acc[3] += av * W[k * 4 + 3];
  }
#pragma unroll
  for (int j = 0; j < 4; j++) Hout[(long)nrow * 4 + j] = acc[j];
}

// asc[n,h] = sum_c h[n,h,c]*a_src[h,c]; adc likewise.
template <int H, int C>
__global__ void gat_attn_coef(const float* __restrict__ Hf,
                              const float* __restrict__ as_, const float* __restrict__ ad_,
                              float* __restrict__ asc, float* __restrict__ adc, int Nn) {
  int nrow = blockIdx.x * blockDim.x + threadIdx.x;
  if (nrow >= Nn) return;
  const float* hr = &Hf[(long)nrow * H * C];
#pragma unroll
  for (int h = 0; h < H; h++) {
    float s = 0.f, d = 0.f;
#pragma unroll
    for (int c = 0; c < C; c++) {
      float v = hr[h * C + c];
      s += v * as_[h * C + c];
      d += v * ad_[h * C + c];
    }
    asc[nrow * H + h] = s;
    adc[nrow * H + h] = d;
  }
}

__global__ void gat_fill(float* __restrict__ p, float v, int count) {
  int i = blockIdx.x * blockDim.x + threadIdx.x;
  if (i < count) p[i] = v;
}

// Pass A: segment max of leaky(asc[src]+adc[dst]) over dst.
template <int H>
__global__ void gat_edge_max(const int* __restrict__ ei,
                             const float* __restrict__ asc, const float* __restrict__ adc,
                             float* __restrict__ mrow) {
  long t = (long)blockIdx.x * blockDim.x + threadIdx.x;
  if (t >= (long)GAT_ETOT * H) return;
  int e = (int)(t / H), h = (int)(t % H);
  int s, d; edge_nodes(ei, e, s, d);
  float v = gat_leaky(asc[s * H + h] + adc[d * H + h]);
  atomicMaxF(&mrow[d * H + h], v);
}

// Pass B: ex = exp(e - m[dst]); store per-edge numerator; segment-sum into z.
template <int H>
__global__ void gat_edge_exp(const int* __restrict__ ei,
                             const float* __restrict__ asc, const float* __restrict__ adc,
                             const float* __restrict__ mrow,
                             float* __restrict__ ex, float* __restrict__ z) {
  long t = (long)blockIdx.x * blockDim.x + threadIdx.x;
  if (t >= (long)GAT_ETOT * H) return;
  int e = (int)(t / H), h = (int)(t % H);
  int s, d; edge_nodes(ei, e, s, d);
  float v = gat_leaky(asc[s * H + h] + adc[d * H + h]);
  float xv = __expf(v - mrow[d * H + h]);
  ex[(long)e * H + h] = xv;
  atomicAdd(&z[d * H + h], xv);
}

// Pass C: agg[dst] += h[src] * alpha. One thread per (edge, 4 consecutive cols).
template <int H, int C>
__global__ void gat_edge_agg(const int* __restrict__ ei,
                             const float* __restrict__ Hf,
                             const float* __restrict__ ex, const float* __restrict__ z,
                             float* __restrict__ agg) {
  constexpr int HC = H * C;
  constexpr int P = HC >> 2;                   // 4-col chunks per edge
  long t = (long)blockIdx.x * blockDim.x + threadIdx.x;
  if (t >= (long)GAT_ETOT * P) return;
  int e = (int)(t / P), part = (int)(t % P);
  int col0 = part << 2;
  int h = col0 / C;
  int s, d; edge_nodes(ei, e, s, d);
  float alpha = ex[(long)e * H + h] / z[d * H + h];
  const float4 hv = *(const float4*)&Hf[(long)s * HC + col0];
  float* o = &agg[(long)d * HC + col0];
  atomicAdd(&o[0], hv.x * alpha);
  atomicAdd(&o[1], hv.y * alpha);
  atomicAdd(&o[2], hv.z * alpha);
  atomicAdd(&o[3], hv.w * alpha);
}

// next = relu(agg + b)
__global__ void gat_bias_relu(const float* __restrict__ agg, const float* __restrict__ b,
                              float* __restrict__ outF, int Nn, int HC) {
  long i = (long)blockIdx.x * blockDim.x + threadIdx.x;
  if (i >= (long)Nn * HC) return;
  float v = agg[i] + b[i % HC];
  outF[i] = v > 0.f ? v : 0.f;
}

// out = sigmoid(agg + b3) * 100
__global__ void gat_final(const float* __restrict__ agg, const float* __restrict__ b,
                          float* __restrict__ out, int Nn) {
  long i = (long)blockIdx.x * blockDim.x + threadIdx.x;
  if (i >= (long)Nn * 4) return;
  float v = agg[i] + b[i & 3];
  out[i] = 100.0f / (1.0f + __expf(-v));
}

template <int H, int C>
static void run_attention(const float* feat, const float* as_, const float* ad_,
                          const int* ei, float* ASC, float* ADC, float* MM, float* ZZ,
                          float* EX, float* AGG, hipStream_t stream) {
  constexpr int HC = H * C;
  const int TB = 256;
  gat_attn_coef<H, C><<<(GAT_N + TB - 1) / TB, TB, 0, stream>>>(feat, as_, ad_, ASC, ADC, GAT_N);
  gat_fill<<<(GAT_N * H  + TB - 1) / TB, TB, 0, stream>>>(MM, -FLT_MAX, GAT_N * H);
  gat_fill<<<(GAT_N * H  + TB - 1) / TB, TB, 0, stream>>>(ZZ, 0.0f, GAT_N * H);
  gat_fill<<<(GAT_N * HC + TB - 1) / TB, TB, 0, stream>>>(AGG, 0.0f, GAT_N * HC);
  long ce = (long)GAT_ETOT * H;
  gat_edge_max<H><<<(int)((ce + TB - 1) / TB), TB, 0, stream>>>(ei, ASC, ADC, MM);
  gat_edge_exp<H><<<(int)((ce + TB - 1) / TB), TB, 0, stream>>>(ei, ASC, ADC, MM, EX, ZZ);
  long ca = (long)GAT_ETOT * (HC >> 2);
  gat_edge_agg<H, C><<<(int)((ca + TB - 1) / TB), TB, 0, stream>>>(ei, feat, EX, ZZ, AGG);
}

extern "C" void kernel_launch(void* const* d_in, const int* in_sizes, int n_in,
                              void* d_out, int out_size, void* d_ws, size_t ws_size,
                              hipStream_t stream) {
  const float* x   = (const float*)d_in[0];
  const int*   ei  = (const int*)d_in[1];
  const float* W1  = (const float*)d_in[2];
  const float* a1s = (const float*)d_in[3];
  const float* a1d = (const float*)d_in[4];
  const float* b1  = (const float*)d_in[5];
  const float* W2  = (const float*)d_in[6];
  const float* a2s = (const float*)d_in[7];
  const float* a2d = (const float*)d_in[8];
  const float* b2  = (const float*)d_in[9];
  const float* W3  = (const float*)d_in[10];
  const float* a3s = (const float*)d_in[11];
  const float* a3d = (const float*)d_in[12];
  const float* b3  = (const float*)d_in[13];
  float* out = (float*)d_out;

  float* ws  = (float*)d_ws;
  float* F1  = ws;                                  // N x 64  (gemm output h)
  float* F2  = ws + (size_t)GAT_N * 64;             // N x 64  (layer input)
  float* AGG = ws + (size_t)GAT_N * 128;            // N x 64  (aggregation)
  float* ASC = ws + (size_t)GAT_N * 192;            // N x 2
  float* ADC = ASC + (size_t)GAT_N * 2;             // N x 2
  float* MM  = ADC + (size_t)GAT_N * 2;             // N x 2
  float* ZZ  = MM  + (size_t)GAT_N * 2;             // N x 2
  float* EX  = ZZ  + (size_t)GAT_N * 2;             // Etot x 2
  float* XP  = EX  + (size_t)GAT_ETOT * 2;          // N x 16 (padded x)
  float* WP  = XP  + (size_t)GAT_N * 16;            // 16 x 64 (padded W1)

  const int TB = 256;
  const int gemm_blocks = (GAT_N + 127) / 128;      // 8 waves x 16 rows / block

  // ---- Layer 1: x[N,10] (padded to 16) -> h1[N,64] ----
  gat_pad_x<<<(GAT_N * 16 + TB - 1) / TB, TB, 0, stream>>>(x, XP, GAT_N);
  gat_pad_w1<<<4, TB, 0, stream>>>(W1, WP);
  gat_gemm_wmma64<16><<<gemm_blocks, TB, 0, stream>>>(XP, WP, F1, GAT_N);
  run_attention<2, 32>(F1, a1s, a1d, ei, ASC, ADC, MM, ZZ, EX, AGG, stream);
  gat_bias_relu<<<(int)(((long)GAT_N * 64 + TB - 1) / TB), TB, 0, stream>>>(AGG, b1, F2, GAT_N, 64);

  // ---- Layer 2: F2[N,64] -> h2[N,64] ----
  gat_gemm_wmma64<64><<<gemm_blocks, TB, 0, stream>>>(F2, W2, F1, GAT_N);
  run_attention<2, 32>(F1, a2s, a2d, ei, ASC, ADC, MM, ZZ, EX, AGG, stream);
  gat_bias_relu<<<(int)(((long)GAT_N * 64 + TB - 1) / TB), TB, 0, stream>>>(AGG, b2, F2, GAT_N, 64);

  // ---- Layer 3: F2[N,64] -> h3[N,4], heads=1 ----
  gat_gemm3<<<(GAT_N + TB - 1) / TB, TB, 0, stream>>>(F2, W3, F1, GAT_N);
  run_attention<1, 4>(F1, a3s, a3d, ei, ASC, ADC, MM, ZZ, EX, AGG, stream);
  gat_final<<<(int)(((long)GAT_N * 4 + TB - 1) / TB), TB, 0, stream>>>(AGG, b3, out, GAT_N);
}